// TopologicalMap_62921270886777
// MI455X (gfx1250) — compile-verified
//
#include <hip/hip_runtime.h>

typedef __attribute__((ext_vector_type(2))) float v2f;
typedef __attribute__((ext_vector_type(8))) float v8f;

#define B_DIM   2048
#define D_DIM   512
#define N_DIM   16384
#define SIDE    128
#define TILE_K  32
#define LDA     36    // 128x32 A tile, padded stride (16B aligned, conflict-free)
#define LDB     132   // 32x128 B tile, padded stride (16B aligned, conflict-free)

// ---------------------------------------------------------------- x row sums
__global__ __launch_bounds__(256) void som_row_sq(const float* __restrict__ x,
                                                  float* __restrict__ x2) {
    __shared__ float red[256];
    const int m = blockIdx.x, tid = threadIdx.x;
    float a = x[m * D_DIM + tid];
    float b = x[m * D_DIM + 256 + tid];
    red[tid] = a * a + b * b;
    __syncthreads();
    for (int off = 128; off > 0; off >>= 1) {
        if (tid < off) red[tid] += red[tid + off];
        __syncthreads();
    }
    if (tid == 0) x2[m] = red[0];
}

// ---------------------------------------------------------------- W col sums
__global__ __launch_bounds__(256) void som_col_sq(const float* __restrict__ w,
                                                  float* __restrict__ w2) {
    const int n = blockIdx.x * 256 + threadIdx.x;
    float s = 0.f;
    for (int k = 0; k < D_DIM; ++k) {
        float v = w[(size_t)k * N_DIM + n];
        s = fmaf(v, v, s);
    }
    w2[n] = s;
}

// ------------------------------------------------- GEMM: norms2 -> d_out
// norms2[m][n] = max(x2[m] + w2[n] - 2 * (x @ W)[m][n], 0)
__global__ __launch_bounds__(256) void som_gemm(const float* __restrict__ x,
                                                const float* __restrict__ w,
                                                const float* __restrict__ x2,
                                                const float* __restrict__ w2,
                                                float* __restrict__ out) {
    __shared__ float As[128 * LDA];
    __shared__ float Bs[TILE_K * LDB];

    const int tid  = threadIdx.x;
    const int wave = tid >> 5;
    const int lane = tid & 31;
    const int hi   = lane >> 4;      // lane half select
    const int lo   = lane & 15;
    const int m0   = blockIdx.y * 128;
    const int n0   = blockIdx.x * 128;
    const int nloc = wave * 16 + lo; // this wave's N column within the tile

    v8f acc[8];
#pragma unroll
    for (int s = 0; s < 8; ++s)
        acc[s] = (v8f){0.f, 0.f, 0.f, 0.f, 0.f, 0.f, 0.f, 0.f};

    for (int kc = 0; kc < D_DIM; kc += TILE_K) {
        __syncthreads();
        // stage A tile: 128 rows x 32 k  (1024 float4, 4 per thread)
#pragma unroll
        for (int i = 0; i < 4; ++i) {
            int idx = tid + 256 * i;
            int row = idx >> 3;            // 8 float4 per row
            int c4  = (idx & 7) << 2;
            float4 v = *(const float4*)(x + (size_t)(m0 + row) * D_DIM + kc + c4);
            *(float4*)(&As[row * LDA + c4]) = v;
        }
        // stage B tile: 32 k x 128 cols  (1024 float4, 4 per thread)
#pragma unroll
        for (int i = 0; i < 4; ++i) {
            int idx = tid + 256 * i;
            int row = idx >> 5;            // 32 float4 per row
            int c4  = (idx & 31) << 2;
            float4 v = *(const float4*)(w + (size_t)(kc + row) * N_DIM + n0 + c4);
            *(float4*)(&Bs[row * LDB + c4]) = v;
        }
        __syncthreads();

#pragma unroll
        for (int ks = 0; ks < TILE_K; ks += 4) {
            // B fragment: b.x = B[ks+2hi][n], b.y = B[ks+2hi+1][n]
            v2f b;
            b.x = Bs[(ks + 2 * hi) * LDB + nloc];
            b.y = Bs[(ks + 2 * hi + 1) * LDB + nloc];
#pragma unroll
            for (int s = 0; s < 8; ++s) {
                // A fragment: a.x = A[lo][ks+2hi], a.y = A[lo][ks+2hi+1]
                const float* ap = &As[(s * 16 + lo) * LDA + ks + 2 * hi];
                v2f a;
                a.x = ap[0];
                a.y = ap[1];
                acc[s] = __builtin_amdgcn_wmma_f32_16x16x4_f32(
                    false, a, false, b, (short)0, acc[s], false, false);
            }
        }
    }

    // epilogue: squared distances, clamp at 0, write to d_out
    const int   n   = n0 + nloc;
    const float w2n = w2[n];
#pragma unroll
    for (int s = 0; s < 8; ++s) {
#pragma unroll
        for (int e = 0; e < 8; ++e) {
            int   m   = m0 + s * 16 + e + 8 * hi;  // C/D layout: M = vgpr + 8*hi
            float val = fmaf(-2.f, acc[s][e], x2[m] + w2n);
            out[(size_t)m * N_DIM + n] = fmaxf(val, 0.f);
        }
    }
}

// -------------------------------------- BMU argmin + separable phi normalizer
__global__ __launch_bounds__(256) void som_bmu(const float* __restrict__ norms2,
                                               const float* __restrict__ stdp,
                                               float* __restrict__ params) {
    __shared__ float sv[256];
    __shared__ int   si[256];
    __shared__ float ssum[256];
    __shared__ float sums[2];

    const int m = blockIdx.x, tid = threadIdx.x;
    const float* row = norms2 + (size_t)m * N_DIM;

    float best = 3.402823466e38f;
    int   bidx = 0;
    for (int n = tid; n < N_DIM; n += 256) {
        float v = row[n];
        if (v < best) { best = v; bidx = n; }  // n increasing -> keeps first
    }
    sv[tid] = best;
    si[tid] = bidx;
    __syncthreads();
    for (int off = 128; off > 0; off >>= 1) {
        if (tid < off) {
            float ov = sv[tid + off];
            int   oi = si[tid + off];
            if (ov < sv[tid] || (ov == sv[tid] && oi < si[tid])) {
                sv[tid] = ov;
                si[tid] = oi;
            }
        }
        __syncthreads();
    }
    const int   idx    = si[0];
    const float rowf   = (float)(idx >> 7);
    const float colf   = (float)(idx & (SIDE - 1));
    const float stdv   = stdp[0];
    const float inv2s2 = 0.5f / (stdv * stdv);

    // S_r = sum_r exp(-0.5 (r-rowf)^2 / s^2)
    float t = 0.f;
    if (tid < SIDE) {
        float d = (float)tid - rowf;
        t = __expf(-d * d * inv2s2);
    }
    ssum[tid] = t;
    __syncthreads();
    for (int off = 128; off > 0; off >>= 1) {
        if (tid < off) ssum[tid] += ssum[tid + off];
        __syncthreads();
    }
    if (tid == 0) sums[0] = ssum[0];
    __syncthreads();

    // S_c
    t = 0.f;
    if (tid < SIDE) {
        float d = (float)tid - colf;
        t = __expf(-d * d * inv2s2);
    }
    ssum[tid] = t;
    __syncthreads();
    for (int off = 128; off > 0; off >>= 1) {
        if (tid < off) ssum[tid] += ssum[tid + off];
        __syncthreads();
    }
    if (tid == 0) {
        params[m * 4 + 0] = rowf;
        params[m * 4 + 1] = colf;
        params[m * 4 + 2] = 1.f / (sums[0] * ssum[0]);
    }
}

// ---------------------------------- in-place scale: out *= phi / sum(phi)
__global__ __launch_bounds__(256) void som_scale(float* __restrict__ out,
                                                 const float* __restrict__ params,
                                                 const float* __restrict__ stdp) {
    const size_t gid = (size_t)blockIdx.x * 256 + threadIdx.x; // 1 float4 each
    const int m  = (int)(gid >> 12);     // 4096 float4 per row
    const int q  = (int)(gid & 4095);
    const int n  = q << 2;
    const int r  = n >> 7;
    const int c0 = n & (SIDE - 1);

    const float rowf = params[m * 4 + 0];
    const float colf = params[m * 4 + 1];
    const float inv  = params[m * 4 + 2];
    const float stdv = stdp[0];
    const float k    = 0.5f / (stdv * stdv);

    const float dr  = (float)r - rowf;
    const float dr2 = dr * dr;
    const float dc0 = (float)c0 - colf;

    float4 v = ((float4*)out)[gid];
    v.x *= __expf(-(dr2 + dc0 * dc0) * k) * inv;
    float dc1 = dc0 + 1.f;
    v.y *= __expf(-(dr2 + dc1 * dc1) * k) * inv;
    float dc2 = dc0 + 2.f;
    v.z *= __expf(-(dr2 + dc2 * dc2) * k) * inv;
    float dc3 = dc0 + 3.f;
    v.w *= __expf(-(dr2 + dc3 * dc3) * k) * inv;
    ((float4*)out)[gid] = v;
}

// ----------------------------------------------------------------- launcher
extern "C" void kernel_launch(void* const* d_in, const int* in_sizes, int n_in,
                              void* d_out, int out_size, void* d_ws, size_t ws_size,
                              hipStream_t stream) {
    const float* x    = (const float*)d_in[0]; // (2048, 512)
    const float* w    = (const float*)d_in[1]; // (512, 16384)
    const float* stdp = (const float*)d_in[2]; // scalar
    float* out = (float*)d_out;                // (2048, 16384)

    float* ws     = (float*)d_ws;
    float* x2     = ws;                    // 2048
    float* w2     = ws + B_DIM;            // 16384
    float* params = ws + B_DIM + N_DIM;    // 2048 * 4

    som_row_sq<<<B_DIM, 256, 0, stream>>>(x, x2);
    som_col_sq<<<N_DIM / 256, 256, 0, stream>>>(w, w2);
    som_gemm<<<dim3(N_DIM / 128, B_DIM / 128), 256, 0, stream>>>(x, w, x2, w2, out);
    som_bmu<<<B_DIM, 256, 0, stream>>>(out, stdp, params);
    som_scale<<<(B_DIM * (N_DIM / 4)) / 256, 256, 0, stream>>>(out, params, stdp);
}